// GAEncode_53334903882145
// MI455X (gfx1250) — compile-verified
//
#include <hip/hip_runtime.h>

#define Bsz  4
#define Npts 4096
#define Cch  128
#define Kn   32

typedef __attribute__((ext_vector_type(2))) float v2f;
typedef __attribute__((ext_vector_type(8))) float v8f;
typedef __attribute__((ext_vector_type(4))) unsigned int u32x4;
typedef __attribute__((ext_vector_type(8))) int i32x8;
typedef __attribute__((ext_vector_type(4))) int i32x4;

__device__ __forceinline__ float wred_max_f(float v) {
#pragma unroll
    for (int m = 16; m >= 1; m >>= 1) v = fmaxf(v, __shfl_xor(v, m, 32));
    return v;
}
__device__ __forceinline__ float wred_min_f(float v) {
#pragma unroll
    for (int m = 16; m >= 1; m >>= 1) v = fminf(v, __shfl_xor(v, m, 32));
    return v;
}
__device__ __forceinline__ int wred_min_i(int v) {
#pragma unroll
    for (int m = 16; m >= 1; m >>= 1) { int o = __shfl_xor(v, m, 32); v = (o < v) ? o : v; }
    return v;
}

// ---------------------------------------------------------------------------
// Kernel 1: fused pairwise-distance (WMMA f32 16x16x4) + top-32 selection.
// pd'[i,j] = 2*xi.xj - ||xj||^2  (row-constant -||xi||^2 dropped: order-preserving)
// Grid: (N/16, B). Block: 512 threads = 16 waves.
// ---------------------------------------------------------------------------
__global__ __launch_bounds__(512)
void knn_topk_kernel(const float* __restrict__ xyz, int* __restrict__ outIdx) {
    const int b    = blockIdx.y;
    const int rt   = blockIdx.x;           // 16-row tile
    const int tid  = threadIdx.x;
    const int lane = tid & 31;
    const int wv   = tid >> 5;             // 0..15
    const int nl   = lane & 15;
    const int k0   = (lane < 16) ? 0 : 2;  // K offset for A/B operand halves

    __shared__ float sA[16][4];            // A operand rows: {2x,2y,2z,-1}
    __shared__ float sBc[4][512];          // B operand cols for chunk: {x,y,z,xx}
    __shared__ float sD[16][516];          // distance chunk, pitch 516 (8*516%64==32)

    const float* xb = xyz + (size_t)b * 3 * Npts;

    if (tid < 64) {
        int m = tid >> 2, k = tid & 3;
        int irow = rt * 16 + m;
        sA[m][k] = (k < 3) ? 2.0f * xb[k * Npts + irow] : -1.0f;
    }

    float setv = -3.0e38f;                 // running top-32: one (val,idx) per lane
    int   seti = 0;
    float smin = -3.0e38f;                 // cached wave-uniform min of the set

    for (int chunk = 0; chunk < 8; ++chunk) {
        const int cb = chunk * 512;
        {   // stage B columns for this chunk
            int gcol = cb + tid;
            float x = xb[0 * Npts + gcol];
            float y = xb[1 * Npts + gcol];
            float z = xb[2 * Npts + gcol];
            sBc[0][tid] = x; sBc[1][tid] = y; sBc[2][tid] = z;
            sBc[3][tid] = x * x + y * y + z * z;
        }
        __syncthreads();

        {   // distance tiles: wave wv -> col-tiles wv and wv+16
            v2f a;
            a[0] = sA[nl][k0];
            a[1] = sA[nl][k0 + 1];
#pragma unroll
            for (int tt = 0; tt < 2; ++tt) {
                int t  = wv + tt * 16;
                int c0 = t * 16 + nl;
                v2f bb;
                bb[0] = sBc[k0][c0];
                bb[1] = sBc[k0 + 1][c0];
                v8f d = {};
                d = __builtin_amdgcn_wmma_f32_16x16x4_f32(
                        false, a, false, bb, (short)0, d, false, false);
                const int mbase = (lane < 16) ? 0 : 8;
#pragma unroll
                for (int r = 0; r < 8; ++r)
                    sD[mbase + r][c0] = d[r];
            }
        }
        __syncthreads();

        {   // selection: wave wv owns row wv
#pragma unroll 1
            for (int g = 0; g < 16; ++g) {
                float cand = sD[wv][g * 32 + lane];
                int   ccol = cb + g * 32 + lane;
                while (true) {
                    float gmax = wred_max_f(cand);
                    if (!(gmax > smin)) break;       // fast path: 5 bpermutes/group
                    int src = wred_min_i((cand == gmax) ? lane : 64);
                    int dst = wred_min_i((setv == smin) ? lane : 64);
                    int gcol = __shfl(ccol, src, 32);
                    if (lane == src) cand = -3.0e38f;
                    if (lane == dst) { setv = gmax; seti = gcol; }
                    smin = wred_min_f(setv);         // refresh cached min
                }
            }
        }
        __syncthreads();
    }

    int row = rt * 16 + wv;
    outIdx[((size_t)b * Npts + row) * Kn + lane] = seti;
}

// ---------------------------------------------------------------------------
// Kernel 2: gather + mean pool + Linear (WMMA f32 16x16x4).
// W (64KB) is staged into LDS by one TDM tensor_load_to_lds per workgroup,
// asynchronously overlapped with the gather phase; TDM pad feature writes a
// conflict-free 132-dword row pitch. Dynamic LDS layout (bytes):
//   sW   @ 0      : 128 rows * 132 floats = 67584
//   sP   @ 67584  : 16 * 132 floats       =  8448
//   sIdx @ 76032  : 16 * 32 ints          =  2048   (total 78080)
// Grid: (N/16, B). Block: 128 threads = 4 waves.
// ---------------------------------------------------------------------------
#define K2_SW_OFF   0
#define K2_SP_OFF   67584
#define K2_SI_OFF   76032
#define K2_LDS_SZ   78080

__device__ __forceinline__ void tdm_load_W(const float* W, unsigned int ldsOff) {
    unsigned long long ga = (unsigned long long)(uintptr_t)W;
    u32x4 g0;
    g0[0] = 1u;                                            // count=1 (valid), user mode
    g0[1] = ldsOff;                                        // lds_addr (bytes)
    g0[2] = (unsigned int)(ga & 0xFFFFFFFFu);              // global_addr[31:0]
    g0[3] = (unsigned int)((ga >> 32) & 0x01FFFFFFu)       // global_addr[56:32]
          | (2u << 30);                                    // type=2 (image)
    i32x8 g1;
    g1[0] = 0x07920000;            // data_size=4B | pad_enable | interval=128dw | amount=4dw
    g1[1] = (int)(128u << 16);     // tensor_dim0 = 128
    g1[2] = (int)(128u << 16);     // tensor_dim1 = 128
    g1[3] = (int)(128u << 16);     // tile_dim0   = 128
    g1[4] = 128;                   // tile_dim1 = 128, tile_dim2 = 0
    g1[5] = 128;                   // tensor_dim0_stride = 128
    g1[6] = 0;
    g1[7] = 0;
    i32x4 z4 = {0, 0, 0, 0};
#if __clang_major__ >= 23
    i32x8 z8 = {0, 0, 0, 0, 0, 0, 0, 0};
    __builtin_amdgcn_tensor_load_to_lds(g0, g1, z4, z4, z8, 0);
#else
    __builtin_amdgcn_tensor_load_to_lds(g0, g1, z4, z4, 0);
#endif
}

__global__ __launch_bounds__(128)
void gather_linear_kernel(const float* __restrict__ feats, const float* __restrict__ W,
                          const int* __restrict__ idx, float* __restrict__ out) {
    extern __shared__ char smem[];
    float* sW   = (float*)(smem + K2_SW_OFF);   // [128][132]
    float* sP   = (float*)(smem + K2_SP_OFF);   // [16][132]
    int*   sIdx = (int*)  (smem + K2_SI_OFF);   // [16][32]

    const int b    = blockIdx.y;
    const int rt   = blockIdx.x;
    const int tid  = threadIdx.x;
    const int lane = tid & 31;
    const int wv   = tid >> 5;
    const int nl   = lane & 15;
    const int k0   = (lane < 16) ? 0 : 2;

    // kick off async TDM copy of W into LDS (wave 0 only; overlaps gather)
    if (wv == 0) {
        tdm_load_W(W, (unsigned int)(uintptr_t)sW + 0u);
    }

    const int rowbase = rt * 16;
    for (int i = tid; i < 16 * 32; i += 128)
        sIdx[i] = idx[((size_t)b * Npts + rowbase + (i >> 5)) * Kn + (i & 31)];

    const float* fb = feats + (size_t)b * Npts * Cch;
    __builtin_prefetch((const void*)(fb + (size_t)tid * Cch), 0, 1);  // global_prefetch_b8
    __syncthreads();

#pragma unroll 1
    for (int m = 0; m < 16; ++m) {           // thread tid == channel c
        float acc = 0.0f;
#pragma unroll
        for (int k = 0; k < Kn; ++k)
            acc += fb[(size_t)sIdx[m * 32 + k] * Cch + tid];   // 512B coalesced per (m,k)
        sP[m * 132 + tid] = acc * (1.0f / Kn);
    }

    // drain the TDM transfer, then publish to all waves
    if (wv == 0) {
        __builtin_amdgcn_s_wait_tensorcnt(0);
    }
    __syncthreads();

#pragma unroll 1
    for (int dt = wv; dt < 8; dt += 4) {
        const int dbase = dt * 16;
        const float* wrow = sW + (size_t)(dbase + nl) * 132;   // B[k,n] = W[dbase+n][k]
        v8f acc = {};
#pragma unroll 1
        for (int kt = 0; kt < 32; ++kt) {
            v2f a, bb;
            a[0]  = sP[nl * 132 + 4 * kt + k0];
            a[1]  = sP[nl * 132 + 4 * kt + k0 + 1];
            bb[0] = wrow[4 * kt + k0];
            bb[1] = wrow[4 * kt + k0 + 1];
            acc = __builtin_amdgcn_wmma_f32_16x16x4_f32(
                      false, a, false, bb, (short)0, acc, false, false);
        }
        const int mbase = (lane < 16) ? 0 : 8;
        float* orow = out + ((size_t)b * Npts + rowbase) * Cch;
#pragma unroll
        for (int r = 0; r < 8; ++r)
            orow[(size_t)(mbase + r) * Cch + dbase + nl] = acc[r];
    }
}

extern "C" void kernel_launch(void* const* d_in, const int* in_sizes, int n_in,
                              void* d_out, int out_size, void* d_ws, size_t ws_size,
                              hipStream_t stream) {
    const float* xyz   = (const float*)d_in[0];   // [B,3,N]
    const float* feats = (const float*)d_in[1];   // [B,N,C]
    const float* W     = (const float*)d_in[2];   // [C,C]
    float* out = (float*)d_out;                   // [B,N,C] fp32
    int*   nbr = (int*)d_ws;                      // [B,N,K] int32 (2 MB)

    dim3 g1(Npts / 16, Bsz), b1(512);
    knn_topk_kernel<<<g1, b1, 0, stream>>>(xyz, nbr);

    dim3 g2(Npts / 16, Bsz), b2(128);
    gather_linear_kernel<<<g2, b2, K2_LDS_SZ, stream>>>(feats, W, nbr, out);
}